// QuantumNN_23682449670661
// MI455X (gfx1250) — compile-verified
//
#include <hip/hip_runtime.h>
#include <hip/hip_bf16.h>

typedef __attribute__((ext_vector_type(2))) float v2f;
typedef __attribute__((ext_vector_type(8))) float v8f;

// ---------------- WMMA wrapper: D = A(16x4) * B(4x16) + C, fp32 ----------------
static __device__ __forceinline__ v8f wmma4(v2f a, v2f b, v8f c) {
  return __builtin_amdgcn_wmma_f32_16x16x4_f32(
      /*neg_a=*/false, a, /*neg_b=*/false, b,
      /*c_mod=*/(short)0, c, /*reuse_a=*/false, /*reuse_b=*/false);
}

// ---------------- Kernel 1: build fixed 16x16 unitary U from qw ----------------
template <int W>  // W = wire (qubit index), bit position is 3-W
static __device__ __forceinline__ void apply1q(
    float (&sr)[16], float (&si)[16],
    float g00r, float g00i, float g01r, float g01i,
    float g10r, float g10i, float g11r, float g11i) {
  constexpr int P = 3 - W;
#pragma unroll
  for (int k = 0; k < 16; ++k) {
    if ((k >> P) & 1) continue;
    const int k1 = k | (1 << P);
    float a0r = sr[k], a0i = si[k], a1r = sr[k1], a1i = si[k1];
    sr[k]  = g00r*a0r - g00i*a0i + g01r*a1r - g01i*a1i;
    si[k]  = g00r*a0i + g00i*a0r + g01r*a1i + g01i*a1r;
    sr[k1] = g10r*a0r - g10i*a0i + g11r*a1r - g11i*a1i;
    si[k1] = g10r*a0i + g10i*a0r + g11r*a1i + g11i*a1r;
  }
}

template <int W>
static __device__ __forceinline__ void gate_ry(float (&sr)[16], float (&si)[16], float t) {
  float c = __cosf(t * 0.5f), s = __sinf(t * 0.5f);
  apply1q<W>(sr, si, c, 0.f, -s, 0.f, s, 0.f, c, 0.f);
}
template <int W>
static __device__ __forceinline__ void gate_rz(float (&sr)[16], float (&si)[16], float t) {
  float c = __cosf(t * 0.5f), s = __sinf(t * 0.5f);
  apply1q<W>(sr, si, c, -s, 0.f, 0.f, 0.f, 0.f, c, s);
}
template <int W>
static __device__ __forceinline__ void gate_rx(float (&sr)[16], float (&si)[16], float t) {
  float c = __cosf(t * 0.5f), s = __sinf(t * 0.5f);
  apply1q<W>(sr, si, c, 0.f, 0.f, -s, 0.f, -s, c, 0.f);
}
template <int C, int T>
static __device__ __forceinline__ void gate_cnot(float (&sr)[16], float (&si)[16]) {
  constexpr int cb = 1 << (3 - C), tb = 1 << (3 - T);
#pragma unroll
  for (int k = 0; k < 16; ++k) {
    if ((k & cb) && !(k & tb)) {
      const int k1 = k | tb;
      float tr = sr[k]; sr[k] = sr[k1]; sr[k1] = tr;
      float ti = si[k]; si[k] = si[k1]; si[k1] = ti;
    }
  }
}

__global__ void qnn_prep(const float* __restrict__ qw, float* __restrict__ U) {
  const int j = threadIdx.x;
  if (j >= 16) return;
  float sr[16], si[16];
#pragma unroll
  for (int k = 0; k < 16; ++k) { sr[k] = (k == j) ? 1.f : 0.f; si[k] = 0.f; }
  float w[16];
#pragma unroll
  for (int i = 0; i < 16; ++i) w[i] = qw[i];

#pragma unroll
  for (int l = 0; l < 2; ++l) {
    gate_ry<0>(sr, si, w[4*l + 0]); gate_rz<0>(sr, si, w[4*l + 4]);
    gate_ry<1>(sr, si, w[4*l + 1]); gate_rz<1>(sr, si, w[4*l + 5]);
    gate_ry<2>(sr, si, w[4*l + 2]); gate_rz<2>(sr, si, w[4*l + 6]);
    gate_ry<3>(sr, si, w[4*l + 3]); gate_rz<3>(sr, si, w[4*l + 7]);
    gate_cnot<0,1>(sr, si); gate_cnot<0,2>(sr, si); gate_cnot<0,3>(sr, si);
    gate_cnot<1,2>(sr, si); gate_cnot<1,3>(sr, si); gate_cnot<2,3>(sr, si);
    gate_rx<0>(sr, si, w[4*l + 8]);  gate_rx<1>(sr, si, w[4*l + 9]);
    gate_rx<2>(sr, si, w[4*l + 10]); gate_rx<3>(sr, si, w[4*l + 11]);
  }
#pragma unroll
  for (int k = 0; k < 16; ++k) {
    U[k * 16 + j]       = sr[k];  // Re U[k][j]
    U[256 + k * 16 + j] = si[k];  // Im U[k][j]
  }
}

// select v[r] for r in [0,8) without scratch spill
static __device__ __forceinline__ float sel8(const float (&v)[8], int r) {
  float s = v[0];
  s = (r == 1) ? v[1] : s; s = (r == 2) ? v[2] : s; s = (r == 3) ? v[3] : s;
  s = (r == 4) ? v[4] : s; s = (r == 5) ? v[5] : s; s = (r == 6) ? v[6] : s;
  s = (r == 7) ? v[7] : s;
  return s;
}

// ---------------- Kernel 2: 16-sample tiles through WMMA pipeline ----------------
__global__ __launch_bounds__(256) void qnn_main(
    const float* __restrict__ x, const float* __restrict__ U,
    const float* __restrict__ W1, const float* __restrict__ b1,
    const float* __restrict__ W2, const float* __restrict__ b2,
    const float* __restrict__ W3, const float* __restrict__ b3,
    const float* __restrict__ W4, const float* __restrict__ b4,
    float* __restrict__ out, int nTiles) {
  __shared__ float lds_t[8][16 * 17];  // per-wave padded transpose tile

  const int lane  = threadIdx.x & 31;
  const int wid   = threadIdx.x >> 5;
  const int n     = lane & 15;      // N column (state index / hidden unit / row id)
  const int hi    = lane >> 4;      // lane half
  const int off   = hi * 2;         // K offset inside a 4-wide chunk
  const int mbase = hi * 8;         // D-layout rows held by this lane
  const int r7    = lane & 7;
  float* T = lds_t[wid];

  // ---- hoisted B-operands (per ISA B layout: N across lanes, K split by half) ----
  v2f urT[4], uiT[4], nuiT[4];
#pragma unroll
  for (int c = 0; c < 4; ++c) {
    const int kb = 4 * c + off;
    urT[c][0]  = U[n * 16 + kb];       urT[c][1]  = U[n * 16 + kb + 1];
    uiT[c][0]  = U[256 + n * 16 + kb]; uiT[c][1]  = U[256 + n * 16 + kb + 1];
    nuiT[c][0] = -uiT[c][0];           nuiT[c][1] = -uiT[c][1];
  }
  v2f w2T[8];
#pragma unroll
  for (int c = 0; c < 8; ++c) {
    const int kb = 4 * c + off;
    w2T[c][0] = W2[n * 32 + kb]; w2T[c][1] = W2[n * 32 + kb + 1];
  }
  v2f w3T[4];
#pragma unroll
  for (int c = 0; c < 4; ++c) {
    const int kb = 4 * c + off;
    w3T[c][0] = (n < 8) ? W3[n * 16 + kb] : 0.f;
    w3T[c][1] = (n < 8) ? W3[n * 16 + kb + 1] : 0.f;
  }
  float w1v[16], b1v[16];
#pragma unroll
  for (int c = 0; c < 8; ++c) {
    const int col = 4 * c + off;
    w1v[2*c] = W1[col]; w1v[2*c + 1] = W1[col + 1];
    b1v[2*c] = b1[col]; b1v[2*c + 1] = b1[col + 1];
  }
  const float b2n = b2[n];
  const float b3n = (n < 8) ? b3[n] : 0.f;
  const float w4n = (n < 8) ? W4[n] : 0.f;
  const float b4s = b4[0];
  const float zsn = (n < 8) ? 1.f : -1.f;

  const int waveGlobal = blockIdx.x * (blockDim.x >> 5) + wid;
  const int numWaves   = gridDim.x * (blockDim.x >> 5);

  for (int tile = waveGlobal; tile < nTiles; tile += numWaves) {
    // ---- product state for batch row (tile*16 + n) ----
    const float4 xv = ((const float4*)x)[tile * 16 + n];
    float s0, c0, s1, c1, s2, c2, s3, c3;
    __sincosf(xv.x * 0.5f, &s0, &c0);
    __sincosf(xv.y * 0.5f, &s1, &c1);
    __sincosf(xv.z * 0.5f, &s2, &c2);
    __sincosf(xv.w * 0.5f, &s3, &c3);
    // u_i = (c^2 - i*c*s , s^2 - i*c*s)
    const float u0r[2] = {c0*c0, s0*s0}, u1r[2] = {c1*c1, s1*s1};
    const float u2r[2] = {c2*c2, s2*s2}, u3r[2] = {c3*c3, s3*s3};
    const float u0i = -c0*s0, u1i = -c1*s1, u2i = -c2*s2, u3i = -c3*s3;
    float qpr[4], qpi[4], ppr[4], ppi[4];
#pragma unroll
    for (int a = 0; a < 2; ++a)
#pragma unroll
      for (int b = 0; b < 2; ++b) {
        qpr[2*a+b] = u0r[a]*u1r[b] - u0i*u1i;
        qpi[2*a+b] = u0r[a]*u1i   + u0i*u1r[b];
        ppr[2*a+b] = u2r[a]*u3r[b] - u2i*u3i;
        ppi[2*a+b] = u2r[a]*u3i   + u2i*u3r[b];
      }
    const float psr0 = hi ? ppr[2] : ppr[0], psi0 = hi ? ppi[2] : ppi[0];
    const float psr1 = hi ? ppr[3] : ppr[1], psi1 = hi ? ppi[3] : ppi[1];

    // ---- statevector: W = V * U^T (complex), 16 x V_WMMA_F32_16X16X4_F32 ----
    v8f Wr = {}, Wi = {};
#pragma unroll
    for (int c = 0; c < 4; ++c) {
      v2f Ar, Ai;
      Ar[0] = qpr[c]*psr0 - qpi[c]*psi0;  Ai[0] = qpr[c]*psi0 + qpi[c]*psr0;
      Ar[1] = qpr[c]*psr1 - qpi[c]*psi1;  Ai[1] = qpr[c]*psi1 + qpi[c]*psr1;
      Wr = wmma4(Ar, urT[c],  Wr);
      Wr = wmma4(Ai, nuiT[c], Wr);
      Wi = wmma4(Ar, uiT[c],  Wi);
      Wi = wmma4(Ai, urT[c],  Wi);
    }

    // ---- q_b = sum_k z_k |w_bk|^2 : reduce over N (lane halves) ----
    float tred[8];
#pragma unroll
    for (int m = 0; m < 8; ++m) {
      float t = (Wr[m]*Wr[m] + Wi[m]*Wi[m]) * zsn;
      t += __shfl_xor(t, 1); t += __shfl_xor(t, 2);
      t += __shfl_xor(t, 4); t += __shfl_xor(t, 8);
      tred[m] = t;
    }
    const float qpick = sel8(tred, r7);               // q for row mbase + r7
    const int   src   = ((n >= 8) ? 16 : 0) + r7;     // lane holding q_n
    const float qrow  = __shfl(qpick, src);           // every lane: q for row n

    // ---- MLP layer1 (per-lane) + layer2 as WMMA: H2 = relu(H1 W2^T + b2) ----
    v8f H2 = {};
#pragma unroll
    for (int c = 0; c < 8; ++c) {
      v2f A;
      A[0] = fmaxf(qrow * w1v[2*c]     + b1v[2*c],     0.f);
      A[1] = fmaxf(qrow * w1v[2*c + 1] + b1v[2*c + 1], 0.f);
      H2 = wmma4(A, w2T[c], H2);
    }
    // D-layout -> A-layout via per-wave LDS tile
#pragma unroll
    for (int m = 0; m < 8; ++m)
      T[(mbase + m) * 17 + n] = fmaxf(H2[m] + b2n, 0.f);
    __builtin_amdgcn_wave_barrier();
    asm volatile("" ::: "memory");

    // ---- layer3 as WMMA: H3 = H2 W3^T (N padded to 16 with zeros) ----
    v8f H3 = {};
#pragma unroll
    for (int c = 0; c < 4; ++c) {
      const int kb = 4 * c + off;
      v2f A;
      A[0] = T[n * 17 + kb];
      A[1] = T[n * 17 + kb + 1];
      H3 = wmma4(A, w3T[c], H3);
    }
    asm volatile("" ::: "memory");

    // ---- layer4: 8-wide dot across lanes + sigmoid ----
    float sred[8];
#pragma unroll
    for (int m = 0; m < 8; ++m) {
      float t = w4n * fmaxf(H3[m] + b3n, 0.f);
      t += __shfl_xor(t, 1); t += __shfl_xor(t, 2);
      t += __shfl_xor(t, 4); t += __shfl_xor(t, 8);
      sred[m] = t;
    }
    if (n < 8) {
      const float z = sel8(sred, r7) + b4s;
      out[tile * 16 + mbase + n] = 1.f / (1.f + __expf(-z));
    }
  }
}

// ---------------- host side ----------------
extern "C" void kernel_launch(void* const* d_in, const int* in_sizes, int n_in,
                              void* d_out, int out_size, void* d_ws, size_t ws_size,
                              hipStream_t stream) {
  const float* x  = (const float*)d_in[0];
  const float* qw = (const float*)d_in[1];
  const float* W1 = (const float*)d_in[2];
  const float* b1 = (const float*)d_in[3];
  const float* W2 = (const float*)d_in[4];
  const float* b2 = (const float*)d_in[5];
  const float* W3 = (const float*)d_in[6];
  const float* b3 = (const float*)d_in[7];
  const float* W4 = (const float*)d_in[8];
  const float* b4 = (const float*)d_in[9];
  float* out = (float*)d_out;
  float* U   = (float*)d_ws;  // Ur[256] then Ui[256]

  const int B      = in_sizes[0] / 4;
  const int nTiles = B / 16;

  qnn_prep<<<1, 16, 0, stream>>>(qw, U);
  qnn_main<<<256, 256, 0, stream>>>(x, U, W1, b1, W2, b2, W3, b3, W4, b4,
                                    out, nTiles);
}